// IMPRLNet_classify_1657857376753
// MI455X (gfx1250) — compile-verified
//
#include <hip/hip_runtime.h>
#include <math.h>

#define NROWS 12000
#define NC    64
#define NEGS  0.01f

// LDS double-buffer staging of packed Z
#define KC        24                   // k-steps (of K=4) per chunk
#define NCH       125                  // 3000 / KC
#define BUF_F     (KC * 256)           // floats per buffer = KC*128*2 = 6144 (24 KB)

typedef __attribute__((ext_vector_type(2))) float v2f;
typedef __attribute__((ext_vector_type(4))) float v4f;
typedef __attribute__((ext_vector_type(8))) float v8f;
typedef __attribute__((ext_vector_type(4))) int   v4i;

#if defined(__AMDGCN__) && __has_builtin(__builtin_amdgcn_global_load_async_to_lds_b128)
#define USE_ASYNC_LDS 1
typedef __attribute__((address_space(1))) v4i* as1_v4i_p;
typedef __attribute__((address_space(3))) v4i* as3_v4i_p;
#endif

__device__ __forceinline__ void wait_async_lds() {
#if defined(__AMDGCN__)
#if __has_builtin(__builtin_amdgcn_s_wait_asynccnt)
  __builtin_amdgcn_s_wait_asynccnt(0);
#else
  asm volatile("s_wait_asynccnt 0" ::: "memory");
#endif
#endif
}

__device__ __forceinline__ float leaky01(float v) { return v >= 0.0f ? v : NEGS * v; }
__device__ __forceinline__ float act_sh(float x, float inv) {
  return leaky01(x - inv) - leaky01(-x - inv);
}

// ---------------- zero small buffer ----------------
__global__ void zero_kernel(float* __restrict__ p, int n) {
  int i = blockIdx.x * blockDim.x + threadIdx.x;
  if (i < n) p[i] = 0.0f;
}

// ---------------- Z0 = act(sum_j Xj Wj^T), Fsum = sum_j Xj Uj^T ----------------
__global__ __launch_bounds__(256) void fused_init_kernel(
    const float* __restrict__ X0, const float* __restrict__ X1, const float* __restrict__ X2,
    const float* __restrict__ W0, const float* __restrict__ W1, const float* __restrict__ W2,
    const float* __restrict__ U0, const float* __restrict__ U1, const float* __restrict__ U2,
    const float* __restrict__ theta,
    float* __restrict__ Z, float* __restrict__ Fs) {
  const int c   = threadIdx.x & 63;
  const int row = blockIdx.x * 4 + (threadIdx.x >> 6);
  const float* Xs[3] = {X0, X1, X2};
  const float* Ws[3] = {W0, W1, W2};
  const float* Us[3] = {U0, U1, U2};
  const int    ds[3] = {512, 768, 1024};
  float accw = 0.0f, accu = 0.0f;
#pragma unroll
  for (int j = 0; j < 3; ++j) {
    const int d = ds[j];
    const float* x = Xs[j] + (size_t)row * d;
    const float* w = Ws[j] + (size_t)c * d;
    const float* u = Us[j] + (size_t)c * d;
    for (int k = 0; k < d; k += 4) {
      v4f xv = *(const v4f*)(x + k);
      v4f wv = *(const v4f*)(w + k);
      v4f uv = *(const v4f*)(u + k);
      accw = fmaf(xv.x, wv.x, accw); accw = fmaf(xv.y, wv.y, accw);
      accw = fmaf(xv.z, wv.z, accw); accw = fmaf(xv.w, wv.w, accw);
      accu = fmaf(xv.x, uv.x, accu); accu = fmaf(xv.y, uv.y, accu);
      accu = fmaf(xv.z, uv.z, accu); accu = fmaf(xv.w, uv.w, accu);
    }
  }
  const float inv = 1.0f / theta[0];
  Z [(size_t)row * NC + c] = act_sh(accw, inv);
  Fs[(size_t)row * NC + c] = accu;
}

// ---------------- partial Gram g += Z^T Z over a 128-row chunk ----------------
__global__ __launch_bounds__(256) void gram_kernel(const float* __restrict__ Z,
                                                   float* __restrict__ g) {
  __shared__ float zs[128 * NC];
  const int base = blockIdx.x * 128;
  for (int i = threadIdx.x; i < 128 * NC; i += 256) {
    int r = base + (i >> 6);
    zs[i] = (r < NROWS) ? Z[(size_t)r * NC + (i & 63)] : 0.0f;
  }
  __syncthreads();
  float acc[16];
#pragma unroll
  for (int i = 0; i < 16; ++i) acc[i] = 0.0f;
  const int c1  = threadIdx.x >> 2;
  const int c2b = (threadIdx.x & 3) * 16;
  for (int r = 0; r < 128; ++r) {
    const float* zr = zs + r * NC;
    const float a = zr[c1];
#pragma unroll
    for (int i = 0; i < 16; ++i) acc[i] = fmaf(a, zr[c2b + i], acc[i]);
  }
#pragma unroll
  for (int i = 0; i < 16; ++i) atomicAdd(&g[c1 * NC + c2b + i], acc[i]);
}

// ---------------- 1/l_norm = 1/sqrt(sum g^2) ----------------
__global__ __launch_bounds__(256) void lnorm_kernel(const float* __restrict__ g,
                                                    float* __restrict__ scal) {
  __shared__ float red[256];
  float s = 0.0f;
  for (int i = threadIdx.x; i < NC * NC; i += 256) { float v = g[i]; s = fmaf(v, v, s); }
  red[threadIdx.x] = s;
  __syncthreads();
  for (int w = 128; w > 0; w >>= 1) {
    if (threadIdx.x < w) red[threadIdx.x] += red[threadIdx.x + w];
    __syncthreads();
  }
  if (threadIdx.x == 0) scal[0] = 1.0f / sqrtf(red[0]);
}

// ---------------- A_sum = adj0 + adj1 + adj2 (optional, if ws fits) ----------------
__global__ __launch_bounds__(256) void adjsum_kernel(const float* __restrict__ a0,
                                                     const float* __restrict__ a1,
                                                     const float* __restrict__ a2,
                                                     float* __restrict__ as) {
  const size_t total  = (size_t)NROWS * NROWS / 4;
  const size_t stride = (size_t)gridDim.x * blockDim.x;
  for (size_t i = (size_t)blockIdx.x * blockDim.x + threadIdx.x; i < total; i += stride) {
    v4f x = __builtin_nontemporal_load(((const v4f*)a0) + i);
    v4f y = __builtin_nontemporal_load(((const v4f*)a1) + i);
    v4f z = __builtin_nontemporal_load(((const v4f*)a2) + i);
    ((v4f*)as)[i] = x + y + z;
  }
}

// ---------------- pack Z into WMMA B-fragment order ----------------
// v2f index = (s*32 + lane)*4 + t  -> lane's 4 tile-fragments are 32 contiguous bytes.
// lane<16:  (Z[4s+0][16t+l], Z[4s+1][16t+l]) ; lane>=16: (Z[4s+2][..], Z[4s+3][..])
__global__ __launch_bounds__(256) void pack_kernel(const float* __restrict__ Zsrc,
                                                   float* __restrict__ Zpack) {
  int idx = blockIdx.x * 256 + threadIdx.x;
  if (idx >= (NROWS / 4) * 128) return;
  int t = idx & 3;
  int l = (idx >> 2) & 31;
  int s = idx >> 7;
  int col = t * 16 + (l & 15);
  int r0  = s * 4 + ((l >> 4) ? 2 : 0);
  v2f v;
  v.x = Zsrc[(size_t)r0 * NC + col];
  v.y = Zsrc[(size_t)(r0 + 1) * NC + col];
  ((v2f*)Zpack)[idx] = v;
}

// ---------------- main diffusion: out = act(((A@Z)/l + Fsum)/3) via WMMA ----------------
template <bool PRESUM>
__global__ __launch_bounds__(256) void diffuse_kernel(
    const float* __restrict__ A0, const float* __restrict__ A1, const float* __restrict__ A2,
    const float* __restrict__ Zpack, const float* __restrict__ Fs,
    const float* __restrict__ scal, const float* __restrict__ theta,
    float* __restrict__ outZ) {
  __shared__ __align__(16) float lds[2 * BUF_F];   // 2 x 24 KB double buffer
  const int tid  = threadIdx.x;
  const int lane = tid & 31;
  const int wave = tid >> 5;
  int tile = blockIdx.x * 8 + wave;
  // Clamp instead of predicating: surplus waves redo the last tile and write
  // identical values to identical addresses (deterministic, branch-free, no
  // exec masking around the WMMA loop -> no accumulator copy chains).
  if (tile >= NROWS / 16) tile = NROWS / 16 - 1;
  const int r0   = tile * 16;
  const int arow = r0 + (lane & 15);                // A frag: lanes 0-15 K0..1, lanes 16-31 K2..3
  const int aoff = (lane >> 4) * 2;
  const float* a0p = A0 + (size_t)arow * NROWS + aoff;
  const float* a1p = A1 + (size_t)arow * NROWS + aoff;
  const float* a2p = A2 + (size_t)arow * NROWS + aoff;
  const v4f* zp4 = (const v4f*)Zpack;               // chunk c starts at v4f index c*KC*64

  v8f acc0 = {0,0,0,0,0,0,0,0};
  v8f acc1 = {0,0,0,0,0,0,0,0};
  v8f acc2 = {0,0,0,0,0,0,0,0};
  v8f acc3 = {0,0,0,0,0,0,0,0};

  // ---- prologue: stage chunk 0 into buffer 0 ----
  {
    const v4f* src = zp4 + tid;
    v4f*       dst = (v4f*)lds + tid;
#ifdef USE_ASYNC_LDS
#pragma unroll
    for (int i = 0; i < (KC * 64) / 256; ++i)
      __builtin_amdgcn_global_load_async_to_lds_b128(
          (as1_v4i_p)(src + i * 256), (as3_v4i_p)(dst + i * 256), 0, 0);
    wait_async_lds();
#else
    v4f stg[(KC * 64) / 256];
#pragma unroll
    for (int i = 0; i < (KC * 64) / 256; ++i) stg[i] = src[i * 256];
#pragma unroll
    for (int i = 0; i < (KC * 64) / 256; ++i) dst[i * 256] = stg[i];
#endif
  }
  __syncthreads();

  for (int c = 0; c < NCH; ++c) {
    const bool havenext = (c + 1 < NCH);
    // ---- issue staging of chunk c+1 into buffer (c+1)&1 ----
#ifdef USE_ASYNC_LDS
    if (havenext) {
      const v4f* src = zp4 + (size_t)(c + 1) * (KC * 64) + tid;
      v4f*       dst = (v4f*)(lds + ((c + 1) & 1) * BUF_F) + tid;
#pragma unroll
      for (int i = 0; i < (KC * 64) / 256; ++i)
        __builtin_amdgcn_global_load_async_to_lds_b128(
            (as1_v4i_p)(src + i * 256), (as3_v4i_p)(dst + i * 256), 0, 0);
    }
#else
    v4f stg[(KC * 64) / 256];
    if (havenext) {
      const v4f* src = zp4 + (size_t)(c + 1) * (KC * 64) + tid;
#pragma unroll
      for (int i = 0; i < (KC * 64) / 256; ++i) stg[i] = src[i * 256];
    }
#endif

    // ---- consume chunk c from buffer c&1 ----
    {
      const char* lbase = (const char*)lds + (c & 1) * (BUF_F * 4) + lane * 32;
      const int kbase = c * KC * 4;
#pragma unroll 4
      for (int s = 0; s < KC; ++s) {
        v2f a;
        if (PRESUM) {
          a = *(const v2f*)(a0p + kbase + 4 * s);
        } else {
          v2f x = *(const v2f*)(a0p + kbase + 4 * s);
          v2f y = *(const v2f*)(a1p + kbase + 4 * s);
          v2f z = *(const v2f*)(a2p + kbase + 4 * s);
          a = x + y + z;
        }
        const v4f* bp = (const v4f*)(lbase + s * 1024);
        v4f b01 = bp[0];
        v4f b23 = bp[1];
        v2f b0 = {b01.x, b01.y};
        v2f b1 = {b01.z, b01.w};
        v2f b2 = {b23.x, b23.y};
        v2f b3 = {b23.z, b23.w};
        acc0 = __builtin_amdgcn_wmma_f32_16x16x4_f32(false, a, false, b0, (short)0, acc0, false, false);
        acc1 = __builtin_amdgcn_wmma_f32_16x16x4_f32(false, a, false, b1, (short)0, acc1, false, false);
        acc2 = __builtin_amdgcn_wmma_f32_16x16x4_f32(false, a, false, b2, (short)0, acc2, false, false);
        acc3 = __builtin_amdgcn_wmma_f32_16x16x4_f32(false, a, false, b3, (short)0, acc3, false, false);
      }
    }

    // ---- commit staging, then rendezvous ----
#ifdef USE_ASYNC_LDS
    wait_async_lds();
#else
    if (havenext) {
      v4f* dst = (v4f*)(lds + ((c + 1) & 1) * BUF_F) + tid;
#pragma unroll
      for (int i = 0; i < (KC * 64) / 256; ++i) dst[i * 256] = stg[i];
    }
#endif
    __syncthreads();
  }

  // ---- epilogue: D layout lane<16 rows r0+0..7, lane>=16 rows r0+8..15; col = 16t + (lane&15)
  {
    const float invl  = scal[0];
    const float invth = 1.0f / theta[0];
    const int colb = lane & 15;
    const int rowa = (lane >> 4) * 8;
    v8f accs[4] = {acc0, acc1, acc2, acc3};
#pragma unroll
    for (int t = 0; t < 4; ++t) {
#pragma unroll
      for (int v = 0; v < 8; ++v) {
        const int row = r0 + rowa + v;
        const int col = t * 16 + colb;
        const float y  = accs[t][v];
        const float zt = fmaf(y, invl, Fs[(size_t)row * NC + col]) * (1.0f / 3.0f);
        outZ[(size_t)row * NC + col] = act_sh(zt, invth);
      }
    }
  }
}

extern "C" void kernel_launch(void* const* d_in, const int* in_sizes, int n_in,
                              void* d_out, int out_size, void* d_ws, size_t ws_size,
                              hipStream_t stream) {
  const float *f[3], *adj[3], *W[3], *U[3], *theta;
  if (n_in >= 13 && in_sizes[1] == NROWS * 768) {
    // function-argument order: f0,f1,f2, a0,a1,a2, W0..2, U0..2, theta
    for (int j = 0; j < 3; ++j) {
      f[j]   = (const float*)d_in[j];
      adj[j] = (const float*)d_in[3 + j];
      W[j]   = (const float*)d_in[6 + j];
      U[j]   = (const float*)d_in[9 + j];
    }
  } else {
    // setup_inputs() dict order: (f,adj,W,U) per view j, then theta
    for (int j = 0; j < 3; ++j) {
      f[j]   = (const float*)d_in[4 * j + 0];
      adj[j] = (const float*)d_in[4 * j + 1];
      W[j]   = (const float*)d_in[4 * j + 2];
      U[j]   = (const float*)d_in[4 * j + 3];
    }
  }
  theta = (const float*)d_in[12];

  float* Z    = (float*)d_ws;                   // 768000 f
  float* Fs   = Z + NROWS * NC;                 // 768000 f
  float* Zp   = Fs + NROWS * NC;                // 768000 f (B-fragment packed)
  float* g    = Zp + NROWS * NC;                // 4096 f
  float* scal = g + NC * NC;                    // 1 f (padded)
  float* Asum = scal + 64;
  const size_t base_bytes = (size_t)(NROWS * NC * 3 + NC * NC + 64) * sizeof(float);
  const bool presum = ws_size >= base_bytes + (size_t)NROWS * NROWS * sizeof(float);

  float* out = (float*)d_out;

  zero_kernel<<<16, 256, 0, stream>>>(g, NC * NC);
  fused_init_kernel<<<NROWS / 4, 256, 0, stream>>>(f[0], f[1], f[2],
                                                   W[0], W[1], W[2],
                                                   U[0], U[1], U[2],
                                                   theta, Z, Fs);
  gram_kernel<<<(NROWS + 127) / 128, 256, 0, stream>>>(Z, g);
  lnorm_kernel<<<1, 256, 0, stream>>>(g, scal);
  if (presum) adjsum_kernel<<<2048, 256, 0, stream>>>(adj[0], adj[1], adj[2], Asum);
  pack_kernel<<<(NROWS / 4) * 128 / 256, 256, 0, stream>>>(Z, Zp);

  for (int b = 0; b < 3; ++b) {
    float* slab = out + (size_t)b * NROWS * NC;
    if (presum)
      diffuse_kernel<true><<<(NROWS / 16 + 7) / 8, 256, 0, stream>>>(
          Asum, Asum, Asum, Zp, Fs, scal, theta, slab);
    else
      diffuse_kernel<false><<<(NROWS / 16 + 7) / 8, 256, 0, stream>>>(
          adj[0], adj[1], adj[2], Zp, Fs, scal, theta, slab);
    if (b < 2) pack_kernel<<<(NROWS / 4) * 128 / 256, 256, 0, stream>>>(slab, Zp);
  }
}